// DirectProjecter_10230612099897
// MI455X (gfx1250) — compile-verified
//
#include <hip/hip_runtime.h>

#define IMG_H 384
#define IMG_W 1280
#define IMG_HW (IMG_H * IMG_W)

// ---------------------------------------------------------------------------
// Kernel 1: initialize key buffer to sentinel (all-ones = "no hit").
// Sentinel has MSB set; any packed (positive-z-bits << 32 | idx) key is
// smaller under unsigned compare, so atomicMin works directly.
// ---------------------------------------------------------------------------
__global__ void dp_init_keys(unsigned long long* __restrict__ keys, int n) {
    int i = blockIdx.x * blockDim.x + threadIdx.x;
    if (i < n) keys[i] = 0xFFFFFFFFFFFFFFFFull;
}

// ---------------------------------------------------------------------------
// Kernel 2: scatter pass. One thread per point. Packs (z_bits, index) into a
// 64-bit key and does a single GLOBAL_ATOMIC_MIN_U64 per point:
//   - min over z_bits  == per-pixel min depth (positive floats are monotone
//     in their unsigned bit patterns)
//   - ties in z broken by min point index (lexicographic key order)
// This exactly reproduces the reference's two-pass scatter-min semantics in
// one pass. Point rows are streamed once -> non-temporal loads keep the key
// buffer resident in the 192MB L2.
// ---------------------------------------------------------------------------
__global__ void dp_scatter(const float* __restrict__ pts,   // (nb, 4, N), chunk base
                           unsigned long long* __restrict__ keys, // (nb, HW)
                           int N) {
    const int b = blockIdx.y;
    const int i = blockIdx.x * blockDim.x + threadIdx.x;
    if (i >= N) return;

    const float* p = pts + (size_t)b * 4 * N;
    const float uf = __builtin_nontemporal_load(p + i);
    const float vf = __builtin_nontemporal_load(p + (size_t)N + i);
    const float z  = __builtin_nontemporal_load(p + (size_t)2 * N + i);

    // jnp.round == round-half-to-even == v_cvt rne
    const int u = __float2int_rn(uf);
    const int v = __float2int_rn(vf);

    if (u < 0 || u >= IMG_W || v < 0 || v >= IMG_H || !(z > 0.0f)) return;

    const int pix = v * IMG_W + u;
    const unsigned long long key =
        ((unsigned long long)__float_as_uint(z) << 32) | (unsigned int)i;
    atomicMin(keys + (size_t)b * IMG_HW + pix, key);  // global_atomic_min_u64
}

// ---------------------------------------------------------------------------
// Kernel 3: resolve pass. One thread per pixel. Unpack winner, gather colors
// (random gathers stay cached: 12MB color table per batch is L2-resident),
// emit depth / 3-channel image / index. Output stores are non-temporal
// (written once, never re-read).
// ---------------------------------------------------------------------------
__global__ void dp_resolve(const unsigned long long* __restrict__ keys, // (nb, HW)
                           const float* __restrict__ colors,            // (nb, 3, N)
                           int N,
                           float* __restrict__ depth_out,  // (nb, HW)
                           float* __restrict__ img_out,    // (nb, 3, HW)
                           float* __restrict__ idx_out) {  // (nb, HW)
    const int b = blockIdx.y;
    const int p = blockIdx.x * blockDim.x + threadIdx.x;
    if (p >= IMG_HW) return;

    const unsigned long long key = keys[(size_t)b * IMG_HW + p];

    float d = 0.0f, c0 = 0.0f, c1 = 0.0f, c2 = 0.0f, io = -1.0f;
    if (key != 0xFFFFFFFFFFFFFFFFull) {
        const unsigned int idx = (unsigned int)(key & 0xFFFFFFFFu);
        d = __uint_as_float((unsigned int)(key >> 32));  // bit-exact winning z
        const float* col = colors + (size_t)b * 3 * N;
        c0 = col[idx];
        c1 = col[(size_t)N + idx];
        c2 = col[(size_t)2 * N + idx];
        io = (float)idx;  // N < 2^24 -> exact in f32
    }

    const size_t pd = (size_t)b * IMG_HW + p;
    __builtin_nontemporal_store(d,  depth_out + pd);
    __builtin_nontemporal_store(c0, img_out + ((size_t)b * 3 + 0) * IMG_HW + p);
    __builtin_nontemporal_store(c1, img_out + ((size_t)b * 3 + 1) * IMG_HW + p);
    __builtin_nontemporal_store(c2, img_out + ((size_t)b * 3 + 2) * IMG_HW + p);
    __builtin_nontemporal_store(io, idx_out + pd);
}

// ---------------------------------------------------------------------------
// Launch: outputs concatenated flat in return order:
//   depth (B*HW) | img (B*3*HW) | index-as-float (B*HW)
// Key buffer lives in d_ws; batches chunked by available workspace.
// All launches on `stream`, no allocation / sync -> graph-capture safe and
// deterministic (ws re-initialized every call).
// ---------------------------------------------------------------------------
extern "C" void kernel_launch(void* const* d_in, const int* in_sizes, int n_in,
                              void* d_out, int out_size, void* d_ws, size_t ws_size,
                              hipStream_t stream) {
    const float* points = (const float*)d_in[0];  // (B, 4, N)
    const float* colors = (const float*)d_in[1];  // (B, 3, N)
    float* out = (float*)d_out;

    int B = out_size / (IMG_HW * 5);
    if (B < 1) B = 1;
    const int N = in_sizes[0] / (4 * B);

    float* depth_out = out;
    float* img_out   = out + (size_t)B * IMG_HW;
    float* idx_out   = out + (size_t)B * IMG_HW * 4;

    unsigned long long* keys = (unsigned long long*)d_ws;
    const size_t per_batch = (size_t)IMG_HW * sizeof(unsigned long long);
    int bchunk = (ws_size >= per_batch) ? (int)(ws_size / per_batch) : 1;
    if (bchunk > B) bchunk = B;

    const dim3 blk(256);  // 8 wave32 waves per block

    for (int b0 = 0; b0 < B; b0 += bchunk) {
        const int nb = (B - b0 < bchunk) ? (B - b0) : bchunk;

        const int n_init = nb * IMG_HW;
        dp_init_keys<<<dim3((n_init + 255) / 256), blk, 0, stream>>>(keys, n_init);

        dp_scatter<<<dim3((N + 255) / 256, nb), blk, 0, stream>>>(
            points + (size_t)b0 * 4 * N, keys, N);

        dp_resolve<<<dim3((IMG_HW + 255) / 256, nb), blk, 0, stream>>>(
            keys, colors + (size_t)b0 * 3 * N, N,
            depth_out + (size_t)b0 * IMG_HW,
            img_out + (size_t)b0 * 3 * IMG_HW,
            idx_out + (size_t)b0 * IMG_HW);
    }
}